// BYOLMSE_214748364897
// MI455X (gfx1250) — compile-verified
//
#include <hip/hip_runtime.h>
#include <hip/hip_bf16.h>

typedef __attribute__((ext_vector_type(2))) float v2f;
typedef __attribute__((ext_vector_type(8))) float v8f;

#define TEMP_ 0.05f
#define EPS_  1e-8f
#define DDIM  1024
#define SEG   256   // columns per wave segment; 4 segments cover DDIM

// Each block: 8 wave32s = 2 row-tiles (16 rows each) x 4 column segments.
// Per wave: stream float4 (global_load_b128) chunks of the owned row,
// accumulate dot / |a|^2 / |b|^2 partials per lane, fold into the WMMA
// A-matrix layout (16 rows x 4 K-slots striped over the wave), then
// V_WMMA_F32_16X16X4_F32 against an all-ones B collapses K, leaving each
// row's sum replicated across all 16 columns of D (layout-safe).
__global__ __launch_bounds__(256) void byol_rowstats_kernel(
    const float* __restrict__ A, const float* __restrict__ B,
    float* __restrict__ blockPartials)
{
    __shared__ float seglds[2][4][3][16];   // [tile][seg][quantity][row]
    __shared__ float lossLds[32];

    const int tid   = threadIdx.x;
    const int wid   = tid >> 5;
    const int lane  = tid & 31;
    const int t     = wid >> 2;             // tile within block: 0..1
    const int seg   = wid & 3;              // column segment: 0..3
    const int tile  = blockIdx.x * 2 + t;
    const int row   = tile * 16 + (lane & 15);
    const int chalf = (lane >> 4) * 4;      // lanes 0-15: cols 8i.. ; lanes 16-31: cols 8i+4..

    const float4* pa = reinterpret_cast<const float4*>(A + (size_t)row * DDIM);
    const float4* pb = reinterpret_cast<const float4*>(B + (size_t)row * DDIM);

    float4 acc_ab = {0.f, 0.f, 0.f, 0.f};
    float4 acc_aa = {0.f, 0.f, 0.f, 0.f};
    float4 acc_bb = {0.f, 0.f, 0.f, 0.f};

    const int cbase = (seg * SEG + chalf) >> 2;   // index in float4 units
#pragma unroll 4
    for (int i = 0; i < SEG / 8; ++i) {
        const float4 av = pa[cbase + 2 * i];      // global_load_b128
        const float4 bv = pb[cbase + 2 * i];
        acc_ab.x = fmaf(av.x, bv.x, acc_ab.x);
        acc_ab.y = fmaf(av.y, bv.y, acc_ab.y);
        acc_ab.z = fmaf(av.z, bv.z, acc_ab.z);
        acc_ab.w = fmaf(av.w, bv.w, acc_ab.w);
        acc_aa.x = fmaf(av.x, av.x, acc_aa.x);
        acc_aa.y = fmaf(av.y, av.y, acc_aa.y);
        acc_aa.z = fmaf(av.z, av.z, acc_aa.z);
        acc_aa.w = fmaf(av.w, av.w, acc_aa.w);
        acc_bb.x = fmaf(bv.x, bv.x, acc_bb.x);
        acc_bb.y = fmaf(bv.y, bv.y, acc_bb.y);
        acc_bb.z = fmaf(bv.z, bv.z, acc_bb.z);
        acc_bb.w = fmaf(bv.w, bv.w, acc_bb.w);
    }

    // Fold 4 per-lane partials into the 2 A-matrix K-slots this lane owns.
    const v2f a_ab = {acc_ab.x + acc_ab.z, acc_ab.y + acc_ab.w};
    const v2f a_aa = {acc_aa.x + acc_aa.z, acc_aa.y + acc_aa.w};
    const v2f a_bb = {acc_bb.x + acc_bb.z, acc_bb.y + acc_bb.w};

    // Cross-K reduction: D = acc(16x4) x ones(4x16) + 0  =>  D[m][n] = rowsum(m) for all n.
    const v2f ones = {1.f, 1.f};
    const v8f zero = {0.f, 0.f, 0.f, 0.f, 0.f, 0.f, 0.f, 0.f};
    v8f d_ab = __builtin_amdgcn_wmma_f32_16x16x4_f32(false, a_ab, false, ones,
                                                     (short)0, zero, false, false);
    v8f d_aa = __builtin_amdgcn_wmma_f32_16x16x4_f32(false, a_aa, false, ones,
                                                     (short)0, zero, false, false);
    v8f d_bb = __builtin_amdgcn_wmma_f32_16x16x4_f32(false, a_bb, false, ones,
                                                     (short)0, zero, false, false);

    // C/D layout: lanes 0-15 vgpr r -> M=r ; lanes 16-31 vgpr r -> M=8+r.
    // Lane 0 (col 0) carries rows 0..7, lane 16 carries rows 8..15.
    if ((lane & 15) == 0) {
        const int mb = (lane >> 4) * 8;
#pragma unroll
        for (int r = 0; r < 8; ++r) {
            seglds[t][seg][0][mb + r] = d_ab[r];
            seglds[t][seg][1][mb + r] = d_aa[r];
            seglds[t][seg][2][mb + r] = d_bb[r];
        }
    }
    __syncthreads();

    // Combine the 4 column segments per row in a FIXED order (deterministic),
    // compute per-row loss 2 - 2*cos.
    if (tid < 32) {
        const int tt = tid >> 4;
        const int m  = tid & 15;
        float dot = ((seglds[tt][0][0][m] + seglds[tt][1][0][m]) +
                     (seglds[tt][2][0][m] + seglds[tt][3][0][m]));
        float saa = ((seglds[tt][0][1][m] + seglds[tt][1][1][m]) +
                     (seglds[tt][2][1][m] + seglds[tt][3][1][m]));
        float sbb = ((seglds[tt][0][2][m] + seglds[tt][1][2][m]) +
                     (seglds[tt][2][2][m] + seglds[tt][3][2][m]));
        const float n1 = sqrtf(saa);
        const float n2 = sqrtf(sbb);
        const float denom = fmaxf(n1, EPS_) * fmaxf(n2, EPS_);
        const float cosv  = dot / denom;
        lossLds[tid] = 2.0f - 2.0f * cosv;
    }
    __syncthreads();

    if (tid == 0) {
        float s = 0.f;
#pragma unroll
        for (int i = 0; i < 32; ++i) s += lossLds[i];   // fixed order
        blockPartials[blockIdx.x] = s;
    }
}

// Single-block, fixed-pattern tree reduction of per-block partials -> scalar.
__global__ __launch_bounds__(256) void byol_final_reduce(
    const float* __restrict__ partials, int P, float scale,
    float* __restrict__ out)
{
    __shared__ float red[256];
    const int tid = threadIdx.x;
    float s = 0.f;
    for (int i = tid; i < P; i += 256) s += partials[i];  // fixed stride order
    red[tid] = s;
    __syncthreads();
#pragma unroll
    for (int off = 128; off > 0; off >>= 1) {
        if (tid < off) red[tid] += red[tid + off];
        __syncthreads();
    }
    if (tid == 0) out[0] = red[0] * scale;
}

extern "C" void kernel_launch(void* const* d_in, const int* in_sizes, int n_in,
                              void* d_out, int out_size, void* d_ws, size_t ws_size,
                              hipStream_t stream) {
    const float* A = (const float*)d_in[0];   // online_output [N, 1024] fp32
    const float* B = (const float*)d_in[1];   // target_output [N, 1024] fp32
    float* out = (float*)d_out;               // scalar fp32

    const int total   = in_sizes[0];
    const int N       = total / DDIM;         // 16384 for the reference shapes
    const int nblocks = N / 32;               // 32 rows per block

    float* partials = (float*)d_ws;           // nblocks floats (~2 KB)

    byol_rowstats_kernel<<<nblocks, 256, 0, stream>>>(A, B, partials);

    const float scale = 1.0f / ((float)N * TEMP_);
    byol_final_reduce<<<1, 256, 0, stream>>>(partials, nblocks, scale, out);
}